// WeightedGraphTransformerLayer_33088428049201
// MI455X (gfx1250) — compile-verified
//
#include <hip/hip_runtime.h>
#include <math.h>

// Problem constants (match reference)
#define NN 50000
#define NE 800000
#define DM 128
#define NH 8

// Padded LDS row strides (conflict-free fragment reads: bank = (4*l15+k)%64)
#define ST128 132   // for k-length-128 rows (A slabs, W1^T)
#define ST256 260   // for k-length-256 rows (Mid, W2^T)

typedef float v2f __attribute__((ext_vector_type(2)));
typedef float v8f __attribute__((ext_vector_type(8)));

// ---------------------------------------------------------------------------
// fp32 WMMA GEMM: C[M,128] = A[M,128] @ B[128,128] (+bias) (+resid)
// Block = 256 threads (8 waves) -> 32x128 slab. A slab and transposed B both
// staged in LDS; inner loop is pure ds_load_2addr_b64 + v_wmma_f32_16x16x4.
// Static LDS: (32*132 + 128*132)*4 = 84,480 B (~82.5 KB; 3 WGs per 320KB WGP)
// ---------------------------------------------------------------------------
__global__ void __launch_bounds__(256)
gemm128_wmma(const float* __restrict__ A, const float* __restrict__ B,
             const float* __restrict__ bias, const float* __restrict__ resid,
             float* __restrict__ C, int M)
{
    __shared__ float As[32 * ST128];
    __shared__ float Bt[128 * ST128];   // Bt[n][k] = B[k][n]
    const int row0 = blockIdx.x * 32;

    // stage A slab (zero-padded tail), float4 rows
    for (int i = threadIdx.x; i < 32 * 32; i += 256) {
        int r = row0 + (i >> 5);
        float4 v = make_float4(0.f, 0.f, 0.f, 0.f);
        if (r < M) v = ((const float4*)A)[(size_t)r * 32 + (i & 31)];
        *(float4*)&As[(i >> 5) * ST128 + (i & 31) * 4] = v;
    }
    // stage B transposed (coalesced global reads, scattered LDS writes)
    for (int i = threadIdx.x; i < 128 * 32; i += 256) {
        const int k = i >> 5, n = (i & 31) * 4;
        float4 v = ((const float4*)B)[i];
        Bt[(n + 0) * ST128 + k] = v.x;
        Bt[(n + 1) * ST128 + k] = v.y;
        Bt[(n + 2) * ST128 + k] = v.z;
        Bt[(n + 3) * ST128 + k] = v.w;
    }
    __syncthreads();

    const int wave  = threadIdx.x >> 5;
    const int lane  = threadIdx.x & 31;
    const int l15   = lane & 15;
    const int hi    = lane >> 4;
    const int khalf = hi * 2;

    #pragma unroll
    for (int t = 0; t < 2; ++t) {
        const int tile = wave * 2 + t;   // 2 row-tiles x 8 col-tiles
        const int rt = tile >> 3;
        const int ct = tile & 7;
        const float* __restrict__ Ap = &As[(rt * 16 + l15) * ST128];
        const float* __restrict__ Bp = &Bt[(ct * 16 + l15) * ST128];
        v8f acc = {};
        #pragma unroll
        for (int k = 0; k < 128; k += 4) {
            v2f a, b;
            a.x = Ap[k + khalf];
            a.y = Ap[k + khalf + 1];
            b.x = Bp[k + khalf];
            b.y = Bp[k + khalf + 1];
            acc = __builtin_amdgcn_wmma_f32_16x16x4_f32(
                false, a, false, b, (short)0, acc, false, false);
        }
        const int ccol = ct * 16 + l15;
        const float bv = bias ? bias[ccol] : 0.0f;
        #pragma unroll
        for (int r = 0; r < 8; ++r) {
            const int grow = row0 + rt * 16 + hi * 8 + r;
            if (grow < M) {
                const size_t idx = (size_t)grow * 128 + ccol;
                float v = acc[r] + bv;
                if (resid) v += resid[idx];
                C[idx] = v;
            }
        }
    }
}

// ---------------------------------------------------------------------------
// Fused FFN: T = X + ( relu(X @ W1 + b1) @ W2 + b2 )
// X:[M,128], W1:[128,256], W2:[256,128]. Intermediate (Mid) and transposed
// weight halves live in LDS; all WMMA fragments come from LDS.
// Static LDS: (32*132 + 32*260 + 128*132)*4 = 117,760 B (~115 KB; 2 WGs/WGP)
// ---------------------------------------------------------------------------
__global__ void __launch_bounds__(256)
ffn_wmma(const float* __restrict__ X, const float* __restrict__ W1,
         const float* __restrict__ b1, const float* __restrict__ W2,
         const float* __restrict__ b2, float* __restrict__ T, int M)
{
    __shared__ float Xs[32 * ST128];
    __shared__ float Mid[32 * ST256];
    __shared__ float Wt[128 * ST128];   // reused: W1^T halves, then W2^T halves
    const int row0 = blockIdx.x * 32;

    for (int i = threadIdx.x; i < 32 * 32; i += 256) {
        int r = row0 + (i >> 5);
        float4 v = make_float4(0.f, 0.f, 0.f, 0.f);
        if (r < M) v = ((const float4*)X)[(size_t)r * 32 + (i & 31)];
        *(float4*)&Xs[(i >> 5) * ST128 + (i & 31) * 4] = v;
    }
    __syncthreads();

    const int wave  = threadIdx.x >> 5;
    const int lane  = threadIdx.x & 31;
    const int l15   = lane & 15;
    const int hi    = lane >> 4;
    const int khalf = hi * 2;

    // ---- phase 1: Mid[32,256] = relu(Xs @ W1 + b1), two 128-col halves ----
    for (int hh = 0; hh < 2; ++hh) {
        // stage W1^T half: Wt[c][k] = W1[k][hh*128 + c], c in 0..127
        for (int i = threadIdx.x; i < 128 * 32; i += 256) {
            const int k = i >> 5, n = (i & 31) * 4;
            float4 v = ((const float4*)W1)[k * 64 + hh * 32 + (i & 31)];
            Wt[(n + 0) * ST128 + k] = v.x;
            Wt[(n + 1) * ST128 + k] = v.y;
            Wt[(n + 2) * ST128 + k] = v.z;
            Wt[(n + 3) * ST128 + k] = v.w;
        }
        __syncthreads();
        #pragma unroll
        for (int t = 0; t < 2; ++t) {
            const int tile = wave * 2 + t;   // 2 row-tiles x 8 col-tiles
            const int rt = tile >> 3;
            const int ct = tile & 7;
            const float* Ap = &Xs[(rt * 16 + l15) * ST128];
            const float* Bp = &Wt[(ct * 16 + l15) * ST128];
            v8f acc = {};
            #pragma unroll
            for (int k = 0; k < 128; k += 4) {
                v2f a, b;
                a.x = Ap[k + khalf];
                a.y = Ap[k + khalf + 1];
                b.x = Bp[k + khalf];
                b.y = Bp[k + khalf + 1];
                acc = __builtin_amdgcn_wmma_f32_16x16x4_f32(
                    false, a, false, b, (short)0, acc, false, false);
            }
            const int mcol = hh * 128 + ct * 16 + l15;
            const float bv = b1[mcol];
            #pragma unroll
            for (int r = 0; r < 8; ++r) {
                const int mrow = rt * 16 + hi * 8 + r;
                Mid[mrow * ST256 + mcol] = fmaxf(acc[r] + bv, 0.0f);
            }
        }
        __syncthreads();   // Wt reads + Mid writes complete before next stage
    }

    // ---- phase 2: T = Mid @ W2 + b2 + Xs, two 64-col halves ----
    for (int hh = 0; hh < 2; ++hh) {
        // stage W2^T half: Wt[c][k] = W2[k][hh*64 + c], c in 0..63, k in 0..255
        for (int i = threadIdx.x; i < 256 * 16; i += 256) {
            const int k = i >> 4, n = (i & 15) * 4;
            float4 v = ((const float4*)W2)[k * 32 + hh * 16 + (i & 15)];
            Wt[(n + 0) * ST256 + k] = v.x;
            Wt[(n + 1) * ST256 + k] = v.y;
            Wt[(n + 2) * ST256 + k] = v.z;
            Wt[(n + 3) * ST256 + k] = v.w;
        }
        __syncthreads();
        {
            const int rt = wave >> 2;        // 2 row-tiles x 4 col-tiles
            const int ct = wave & 3;
            const float* Ap = &Mid[(rt * 16 + l15) * ST256];
            const float* Bp = &Wt[(ct * 16 + l15) * ST256];
            v8f acc = {};
            #pragma unroll 8
            for (int k = 0; k < 256; k += 4) {
                v2f a, b;
                a.x = Ap[k + khalf];
                a.y = Ap[k + khalf + 1];
                b.x = Bp[k + khalf];
                b.y = Bp[k + khalf + 1];
                acc = __builtin_amdgcn_wmma_f32_16x16x4_f32(
                    false, a, false, b, (short)0, acc, false, false);
            }
            const int ccol = hh * 64 + ct * 16 + l15;
            const float bv = b2[ccol];
            #pragma unroll
            for (int r = 0; r < 8; ++r) {
                const int lrow = rt * 16 + hi * 8 + r;
                const int grow = row0 + lrow;
                if (grow < M)
                    T[(size_t)grow * 128 + ccol] =
                        acc[r] + bv + Xs[lrow * ST128 + ccol];
            }
        }
        __syncthreads();
    }
}

// ---------------------------------------------------------------------------
// Per-(edge,head) attention: score = K[src]*Q[dst]*0.25*pe (written back over
// pe -> e_attn), s = exp(clip(sum)), scatter wV += V[src]*s, z += s via
// L2-resident fp32 atomics (accumulators are 27 MB, fits 192 MB L2).
// ---------------------------------------------------------------------------
__global__ void __launch_bounds__(256)
edge_attn(const float* __restrict__ Qh, const float* __restrict__ Kh,
          const float* __restrict__ Vh, float* __restrict__ PE,
          const int* __restrict__ src, const int* __restrict__ dst,
          float* __restrict__ wV, float* __restrict__ z)
{
    const long gid = (long)blockIdx.x * 256 + threadIdx.x;
    if (gid >= (long)NE * NH) return;
    const int e    = (int)(gid >> 3);
    const int hidx = (int)(gid & 7);
    const int s = src[e], d = dst[e];
    const float4* kp = (const float4*)(Kh + (size_t)s * 128 + hidx * 16);
    const float4* qp = (const float4*)(Qh + (size_t)d * 128 + hidx * 16);
    const float4* vp = (const float4*)(Vh + (size_t)s * 128 + hidx * 16);
    float4* pp = (float4*)(PE + (size_t)e * 128 + hidx * 16);

    float4 vv[4], sc[4];
    float sum = 0.0f;
    #pragma unroll
    for (int i = 0; i < 4; ++i) {
        float4 kv = kp[i], qv = qp[i], pv = pp[i];
        vv[i] = vp[i];
        sc[i].x = kv.x * qv.x * 0.25f * pv.x;
        sc[i].y = kv.y * qv.y * 0.25f * pv.y;
        sc[i].z = kv.z * qv.z * 0.25f * pv.z;
        sc[i].w = kv.w * qv.w * 0.25f * pv.w;
        sum += sc[i].x + sc[i].y + sc[i].z + sc[i].w;
    }
    #pragma unroll
    for (int i = 0; i < 4; ++i) pp[i] = sc[i];

    const float se = __expf(fminf(fmaxf(sum, -5.0f), 5.0f));
    unsafeAtomicAdd(&z[(size_t)d * NH + hidx], se);
    float* wp = wV + (size_t)d * 128 + hidx * 16;
    #pragma unroll
    for (int i = 0; i < 4; ++i) {
        unsafeAtomicAdd(&wp[i * 4 + 0], vv[i].x * se);
        unsafeAtomicAdd(&wp[i * 4 + 1], vv[i].y * se);
        unsafeAtomicAdd(&wp[i * 4 + 2], vv[i].z * se);
        unsafeAtomicAdd(&wp[i * 4 + 3], vv[i].w * se);
    }
}

// h_attn = wV / (z + 1e-6)
__global__ void attn_div(const float* __restrict__ wV, const float* __restrict__ z,
                         float* __restrict__ out)
{
    long i = (long)blockIdx.x * 256 + threadIdx.x;
    if (i >= (long)NN * 128) return;
    int  col = (int)(i & 127);
    long row = i >> 7;
    out[i] = wV[i] / (z[row * NH + (col >> 4)] + 1e-6f);
}

// Column sums / sums of squares for BatchNorm (biased stats).
__global__ void __launch_bounds__(256)
bn_stats(const float* __restrict__ X, float* __restrict__ stats, int M)
{
    const int c    = threadIdx.x & 127;
    const int half = threadIdx.x >> 7;
    const int r0   = blockIdx.x * 512;
    const int rend = min(r0 + 512, M);
    float s = 0.f, ss = 0.f;
    for (int r = r0 + half; r < rend; r += 2) {
        float v = X[(size_t)r * 128 + c];
        s += v; ss += v * v;
    }
    unsafeAtomicAdd(&stats[c], s);
    unsafeAtomicAdd(&stats[128 + c], ss);
}

// stats[256+c] = g*rsqrt(var+eps), stats[384+c] = b - mean*scale
__global__ void bn_finalize(float* __restrict__ stats, const float* __restrict__ g,
                            const float* __restrict__ b, float invM)
{
    int c = threadIdx.x;
    float mean = stats[c] * invM;
    float var  = stats[128 + c] * invM - mean * mean;
    float sc   = g[c] * rsqrtf(var + 1e-5f);
    stats[256 + c] = sc;
    stats[384 + c] = b[c] - mean * sc;
}

__global__ void bn_apply(const float* __restrict__ X, const float* __restrict__ stats,
                         float* __restrict__ Y, long total)
{
    long i = (long)blockIdx.x * 256 + threadIdx.x;
    if (i >= total) return;
    int c = (int)(i & 127);
    Y[i] = X[i] * stats[256 + c] + stats[384 + c];
}

__global__ void zero_f32(float* __restrict__ p, long n)
{
    long i = (long)blockIdx.x * 256 + threadIdx.x;
    if (i < n) p[i] = 0.0f;
}

// ---------------------------------------------------------------------------
extern "C" void kernel_launch(void* const* d_in, const int* in_sizes, int n_in,
                              void* d_out, int out_size, void* d_ws, size_t ws_size,
                              hipStream_t stream)
{
    const float* h      = (const float*)d_in[0];
    const float* e      = (const float*)d_in[1];
    const float* Wq     = (const float*)d_in[2];
    const float* Wk     = (const float*)d_in[3];
    const float* Wv     = (const float*)d_in[4];
    const float* We     = (const float*)d_in[5];
    const float* O_h_w  = (const float*)d_in[6];
    const float* O_h_b  = (const float*)d_in[7];
    const float* O_e_w  = (const float*)d_in[8];
    const float* O_e_b  = (const float*)d_in[9];
    const float* bn1h_g = (const float*)d_in[10];
    const float* bn1h_b = (const float*)d_in[11];
    const float* bn1e_g = (const float*)d_in[12];
    const float* bn1e_b = (const float*)d_in[13];
    const float* fh_w1  = (const float*)d_in[14];
    const float* fh_b1  = (const float*)d_in[15];
    const float* fh_w2  = (const float*)d_in[16];
    const float* fh_b2  = (const float*)d_in[17];
    const float* fe_w1  = (const float*)d_in[18];
    const float* fe_b1  = (const float*)d_in[19];
    const float* fe_w2  = (const float*)d_in[20];
    const float* fe_b2  = (const float*)d_in[21];
    const float* bn2h_g = (const float*)d_in[22];
    const float* bn2h_b = (const float*)d_in[23];
    const float* bn2e_g = (const float*)d_in[24];
    const float* bn2e_b = (const float*)d_in[25];
    const int*   src    = (const int*)d_in[26];
    const int*   dst    = (const int*)d_in[27];

    float* out = (float*)d_out;

    // workspace layout (~949 MB fp32, with in-place reuse)
    float* ws = (float*)d_ws;
    float* Qh = ws;                        // [N,128]  Q  -> later h_attn
    float* Kh = Qh + (size_t)NN * 128;     // [N,128]  K  -> later h2n
    float* Vh = Kh + (size_t)NN * 128;     // [N,128]  V  -> later t_h
    float* H2 = Vh + (size_t)NN * 128;     // [N,128]  h2 pre-BN
    float* WV = H2 + (size_t)NN * 128;     // [N,128]  scatter accumulator
    float* Z  = WV + (size_t)NN * 128;     // [N,8]    scatter accumulator
    float* ST = Z  + (size_t)NN * NH;      // [512]    BN stats/scale/shift
    float* PE = ST + 512;                  // [E,128]  pe -> e_attn -> e2n
    float* E2 = PE + (size_t)NE * 128;     // [E,128]  e2 pre-BN -> t_e

    const int gN = (NN + 31) / 32;
    const int gE = (NE + 31) / 32;

    // 0. zero the scatter accumulators (WV and Z are contiguous)
    {
        long n = (long)NN * (128 + NH);
        zero_f32<<<(int)((n + 255) / 256), 256, 0, stream>>>(WV, n);
    }

    // 1. projections (fp32 WMMA GEMMs)
    gemm128_wmma<<<gN, 256, 0, stream>>>(h, Wq, nullptr, nullptr, Qh, NN);
    gemm128_wmma<<<gN, 256, 0, stream>>>(h, Wk, nullptr, nullptr, Kh, NN);
    gemm128_wmma<<<gN, 256, 0, stream>>>(h, Wv, nullptr, nullptr, Vh, NN);
    gemm128_wmma<<<gE, 256, 0, stream>>>(e, We, nullptr, nullptr, PE, NE);

    // 2. edge attention + scatter (PE overwritten with e_attn)
    edge_attn<<<(int)(((long)NE * NH) / 256), 256, 0, stream>>>(
        Qh, Kh, Vh, PE, src, dst, WV, Z);

    // 3. h_attn = wV / (z + 1e-6)  (into Qh, now free)
    attn_div<<<(int)(((long)NN * 128 + 255) / 256), 256, 0, stream>>>(WV, Z, Qh);

    // 4. O projections with fused bias + residual
    gemm128_wmma<<<gN, 256, 0, stream>>>(Qh, O_h_w, O_h_b, h, H2, NN);
    gemm128_wmma<<<gE, 256, 0, stream>>>(PE, O_e_w, O_e_b, e, E2, NE);

    // 5. BN1 (h): stats over H2 -> normalized h2n into Kh
    zero_f32<<<1, 256, 0, stream>>>(ST, 256);
    bn_stats<<<(NN + 511) / 512, 256, 0, stream>>>(H2, ST, NN);
    bn_finalize<<<1, 128, 0, stream>>>(ST, bn1h_g, bn1h_b, 1.0f / NN);
    bn_apply<<<(int)(((long)NN * 128 + 255) / 256), 256, 0, stream>>>(
        H2, ST, Kh, (long)NN * 128);

    // 6. BN1 (e): stats over E2 -> normalized e2n into PE
    zero_f32<<<1, 256, 0, stream>>>(ST, 256);
    bn_stats<<<(NE + 511) / 512, 256, 0, stream>>>(E2, ST, NE);
    bn_finalize<<<1, 128, 0, stream>>>(ST, bn1e_g, bn1e_b, 1.0f / NE);
    bn_apply<<<(int)(((long)NE * 128 + 255) / 256), 256, 0, stream>>>(
        E2, ST, PE, (long)NE * 128);

    // 7. fused FFN + residual (t = x + ffn(x)); intermediates stay in LDS
    ffn_wmma<<<gN, 256, 0, stream>>>(Kh, fh_w1, fh_b1, fh_w2, fh_b2, Vh, NN);
    ffn_wmma<<<gE, 256, 0, stream>>>(PE, fe_w1, fe_b1, fe_w2, fe_b2, E2, NE);

    // 8. BN2 (h) -> d_out[0 : N*128]
    zero_f32<<<1, 256, 0, stream>>>(ST, 256);
    bn_stats<<<(NN + 511) / 512, 256, 0, stream>>>(Vh, ST, NN);
    bn_finalize<<<1, 128, 0, stream>>>(ST, bn2h_g, bn2h_b, 1.0f / NN);
    bn_apply<<<(int)(((long)NN * 128 + 255) / 256), 256, 0, stream>>>(
        Vh, ST, out, (long)NN * 128);

    // 9. BN2 (e) -> d_out[N*128 : N*128 + E*128]
    zero_f32<<<1, 256, 0, stream>>>(ST, 256);
    bn_stats<<<(NE + 511) / 512, 256, 0, stream>>>(E2, ST, NE);
    bn_finalize<<<1, 128, 0, stream>>>(ST, bn2e_g, bn2e_b, 1.0f / NE);
    bn_apply<<<(int)(((long)NE * 128 + 255) / 256), 256, 0, stream>>>(
        E2, ST, out + (size_t)NN * 128, (long)NE * 128);
}